// RandomCrop_91216515432732
// MI455X (gfx1250) — compile-verified
//
#include <hip/hip_runtime.h>
#include <hip/hip_bf16.h>

#define PAD 4

// Single reflection fold suffices because PAD (4) << dimension (256).
__device__ __forceinline__ int reflect_idx(int i, int n) {
    i = (i < 0) ? -i : i;               // left reflect:  -k -> k
    i = (i >= n) ? (2 * n - 2 - i) : i; // right reflect: n-1+k -> n-1-k
    return i;
}

// Layout (row-major): out[b][c][h][w], B=128, C=3, H=256, W=256.
// One block = 4 consecutive rows of one (b,c) plane.
//   blockIdx.x = bc*(H/4) + hc,  bc = b*C + c   (uniform -> SALU/SMEM)
//   threadIdx.x: [7:6] = row-in-chunk, [5:0] = float4-column
__global__ __launch_bounds__(256) void RandomCrop_91216515432732_kernel(
    const float* __restrict__ x,
    const int*   __restrict__ offs_h,
    const int*   __restrict__ offs_w,
    float*       __restrict__ out)
{
    const int W = 256;
    const int H = 256;

    // Wave-uniform sample decode (stays in SGPRs).
    int bc = blockIdx.x >> 6;          // b*C + c   (H/4 = 64 chunks)
    int hc = blockIdx.x & 63;          // row-chunk within plane
    int b  = bc / 3;                   // C = 3 -> SALU magic divide, once per block

    // Per-sample offsets: scalar SMEM loads (b is uniform).
    int dh = offs_h[b] - PAD;          // in [-4, 4]
    int dw = offs_w[b] - PAD;          // in [-4, 4]

    int h  = (hc << 2) + (threadIdx.x >> 6);  // output row
    int w4 = (threadIdx.x & 63) << 2;         // output column (multiple of 4)

    int ih = reflect_idx(h + dh, H);
    const float* __restrict__ src = x + ((size_t)bc * H + ih) * W;

    int j0 = w4 + dw;
    float4 v;
    // (dw & 3)==0 is wave-uniform; bounds check diverges only at row edges.
    if (((dw & 3) == 0) && (j0 >= 0) && (j0 + 3 < W)) {
        v = *(const float4*)(src + j0);                 // global_load_b128
    } else {
        v.x = src[reflect_idx(j0 + 0, W)];
        v.y = src[reflect_idx(j0 + 1, W)];
        v.z = src[reflect_idx(j0 + 2, W)];
        v.w = src[reflect_idx(j0 + 3, W)];
    }

    // Output float4 index collapses to blockIdx*256 + threadIdx.
    *(float4*)(out + (((size_t)blockIdx.x << 8) + threadIdx.x) * 4) = v; // global_store_b128
}

extern "C" void kernel_launch(void* const* d_in, const int* in_sizes, int n_in,
                              void* d_out, int out_size, void* d_ws, size_t ws_size,
                              hipStream_t stream) {
    const float* x      = (const float*)d_in[0];
    const int*   offs_h = (const int*)d_in[1];
    const int*   offs_w = (const int*)d_in[2];
    float*       out    = (float*)d_out;

    // B*C*(H/4) = 128*3*64 = 24,576 blocks; 256 threads = 8 wave32s each.
    // Each block writes 4 rows * 256 floats = 256 float4 -> covers out exactly.
    int grid  = 128 * 3 * 64;
    int block = 256;

    RandomCrop_91216515432732_kernel<<<grid, block, 0, stream>>>(
        x, offs_h, offs_w, out);
}